// FirstDerivO2_46969762349096
// MI455X (gfx1250) — compile-verified
//
#include <hip/hip_runtime.h>
#include <hip/hip_bf16.h>
#include <stdint.h>

// FirstDerivO2: central difference with replicate padding on (8,1,128,128,128) f32.
// Memory-bound: ~256 MiB traffic -> ~11us floor at 23.3 TB/s. Strategy:
// async global->LDS tile staging (gfx1250 GLOBAL_LOAD_ASYNC_TO_LDS_B128,
// tracked by ASYNCcnt), compute from LDS, vectorized b128 stores.

#define N      128
#define SX     (N * N)     // 16384, stride along axis0
#define SY     N           // stride along axis1
#define VOL    (N * N * N) // 2097152 per batch
#define TY     8           // y-rows per block tile
#define SCALE  50.0f       // 0.5 / dx, dx = 0.01

// LDS tile: 3 x-planes, (TY+2) y-rows (with halo), 128 z
#define ROWS   (TY + 2)            // 10
#define PLANE  (ROWS * N)          // 1280 floats
#define TILE_F (3 * PLANE)         // 3840 floats = 15 KiB

__device__ __forceinline__ int clamp_i(int v, int lo, int hi) {
    return v < lo ? lo : (v > hi ? hi : v);
}

__global__ __launch_bounds__(256)
void first_deriv_o2_kernel(const float* __restrict__ u, float* __restrict__ out) {
    __shared__ float tile[TILE_F];

    const int tid   = threadIdx.x;
    const int bid   = blockIdx.x;
    const int ytile = bid & 15;          // 16 y-tiles of TY=8 rows
    const int x     = (bid >> 4) & 127;  // axis0 index
    const int b     = bid >> 11;         // batch 0..7
    const int y0    = ytile * TY;

    const size_t bbase = (size_t)b << 21; // b * VOL

    // ---- Stage 3 planes (x-1, x, x+1) with y halo into LDS via async b128 ----
    // 960 float4 transfers, <=4 per lane. Replicate clamping folded into addresses.
    const uint32_t lds_base = (uint32_t)(uintptr_t)&tile[0]; // low 32b = LDS offset
    for (int v4 = tid; v4 < TILE_F / 4; v4 += 256) {
        const int L   = v4 << 2;                 // float index in tile
        const int p   = L / PLANE;               // 0..2  (x-1, x, x+1)
        const int rem = L - p * PLANE;
        const int r   = rem >> 7;                // 0..9  (y0-1 .. y0+8)
        const int z   = rem & 127;
        const int xs  = clamp_i(x + p - 1, 0, N - 1);
        const int ys  = clamp_i(y0 + r - 1, 0, N - 1);
        const float* g = u + bbase + ((size_t)xs << 14) + ((size_t)ys << 7) + z;
        const uint32_t loff = lds_base + ((uint32_t)L << 2);
        asm volatile("global_load_async_to_lds_b128 %0, %1, off"
                     :: "v"(loff), "v"(g)
                     : "memory");
    }
    asm volatile("s_wait_asynccnt 0" ::: "memory");
    __syncthreads();

    // ---- Compute: each thread owns 4 consecutive z in one y-row ----
    const int yr = tid >> 5;          // 0..7 (row within tile)
    const int z0 = (tid & 31) << 2;   // 0,4,...,124

    const int crow = PLANE + (yr + 1) * N;       // center plane, center row

    // d2 (axis z): clamp at global z boundaries (tile spans full z)
    const float vm = tile[crow + (z0 == 0 ? 0 : z0 - 1)];
    const float v0 = tile[crow + z0 + 0];
    const float v1 = tile[crow + z0 + 1];
    const float v2 = tile[crow + z0 + 2];
    const float v3 = tile[crow + z0 + 3];
    const float vp = tile[crow + (z0 + 4 > N - 1 ? N - 1 : z0 + 4)];
    float4 d2v;
    d2v.x = (v1 - vm) * SCALE;
    d2v.y = (v2 - v0) * SCALE;
    d2v.z = (v3 - v1) * SCALE;
    d2v.w = (vp - v2) * SCALE;

    // d0 (axis x): plane 2 minus plane 0, same row
    const int rm = 0 * PLANE + (yr + 1) * N + z0;
    const int rp = 2 * PLANE + (yr + 1) * N + z0;
    float4 d0v;
    d0v.x = (tile[rp + 0] - tile[rm + 0]) * SCALE;
    d0v.y = (tile[rp + 1] - tile[rm + 1]) * SCALE;
    d0v.z = (tile[rp + 2] - tile[rm + 2]) * SCALE;
    d0v.w = (tile[rp + 3] - tile[rm + 3]) * SCALE;

    // d1 (axis y): row yr+2 minus row yr in center plane
    const int ym = PLANE + (yr + 0) * N + z0;
    const int yp = PLANE + (yr + 2) * N + z0;
    float4 d1v;
    d1v.x = (tile[yp + 0] - tile[ym + 0]) * SCALE;
    d1v.y = (tile[yp + 1] - tile[ym + 1]) * SCALE;
    d1v.z = (tile[yp + 2] - tile[ym + 2]) * SCALE;
    d1v.w = (tile[yp + 3] - tile[ym + 3]) * SCALE;

    // ---- Vectorized stores into concatenated (d0, d1, d2) output ----
    const size_t o = bbase + ((size_t)x << 14) + ((size_t)(y0 + yr) << 7) + z0;
    *reinterpret_cast<float4*>(out + o)                       = d0v;
    *reinterpret_cast<float4*>(out + (size_t)8 * VOL + o)     = d1v;
    *reinterpret_cast<float4*>(out + (size_t)16 * VOL + o)    = d2v;
}

extern "C" void kernel_launch(void* const* d_in, const int* in_sizes, int n_in,
                              void* d_out, int out_size, void* d_ws, size_t ws_size,
                              hipStream_t stream) {
    (void)in_sizes; (void)n_in; (void)out_size; (void)d_ws; (void)ws_size;
    const float* u = (const float*)d_in[0];
    float* out = (float*)d_out;
    // 8 batches * 128 x-slices * 16 y-tiles = 16384 blocks of 256 threads
    dim3 grid(8 * 128 * 16);
    dim3 block(256);
    first_deriv_o2_kernel<<<grid, block, 0, stream>>>(u, out);
}